// QuantumLSTMCell_65481071400604
// MI455X (gfx1250) — compile-verified
//
#include <hip/hip_runtime.h>
#include <hip/hip_bf16.h>

#define BATCH 16384
#define DIMX  256
#define DIMH  1024
#define DH    1280   // D + H

typedef __attribute__((ext_vector_type(2))) float v2f;
typedef __attribute__((ext_vector_type(8))) float v8f;

#define WMMA_F32(A, B, C) \
  __builtin_amdgcn_wmma_f32_16x16x4_f32(false, (A), false, (B), (short)0, (C), false, false)

// ---------------------------------------------------------------------------
// K1: pre-activations  a_ws[b][gate] = comb[b,:] . W_gate[0,:]   (bias in K2)
//
// V_WMMA_F32_16X16X4_F32, M = 16 batch rows, N = 16 (gates 0..3 live),
// K-loop of 320 steps of 4.  Per ISA layout:
//   A (16x4):  lanes 0-15 -> row=lane,    K={k,k+1}; lanes 16-31 -> row=lane-16, K={k+2,k+3}
//   B (4x16):  lanes 0-15 -> N=lane,      K={k,k+1}; lanes 16-31 -> N=lane-16,   K={k+2,k+3}
//   D:         VGPR v -> row = v + 8*(lane>=16), gate = lane&15
//
// The K loop is split at the x/hx boundary (256) so each phase streams from a
// single base pointer with immediate offsets.  Lanes with N>=4 read a zeroed
// LDS region, so B loads are unconditional (no cndmask in the hot loop).
// ---------------------------------------------------------------------------
__global__ __launch_bounds__(32) void qlstm_gemv_wmma(
    const float* __restrict__ x,  const float* __restrict__ hx,
    const float* __restrict__ Wf, const float* __restrict__ Wi,
    const float* __restrict__ Wg, const float* __restrict__ Wo,
    float* __restrict__ a_ws) {
  __shared__ float lw[2 * DH * 4];             // [0,5120): wp[k][gate]; [5120,10240): zeros
  const int lane = threadIdx.x;
  for (int k = lane; k < DH; k += 32) {        // stage row-0 of each W, packed
    lw[k * 4 + 0] = Wf[k];
    lw[k * 4 + 1] = Wi[k];
    lw[k * 4 + 2] = Wg[k];
    lw[k * 4 + 3] = Wo[k];
  }
  for (int k = lane; k < DH * 4; k += 32)      // zero region for dead N lanes
    lw[DH * 4 + k] = 0.0f;
  __syncthreads();

  const int l15  = lane & 15;
  const int half = lane >> 4;                  // 0 or 1
  const int koff = half * 2;                   // K sub-offset for this half-wave
  const int row  = blockIdx.x * 16 + l15;

  const float* __restrict__ px = x  + (size_t)row * DIMX + koff;
  const float* __restrict__ ph = hx + (size_t)row * DIMH + koff;
  const float* __restrict__ pb = lw + ((l15 < 4) ? 0 : DH * 4) + koff * 4 + (l15 & 3);

  v8f c0 = {}, c1 = {}, c2 = {}, c3 = {};

  // ---- phase 1: j in [0,256) from x ----
  #pragma unroll 2
  for (int k = 0; k < DIMX; k += 16) {
    v2f a, b;
    a = *(const v2f*)(px + k +  0); b.x = pb[(k +  0) * 4]; b.y = pb[(k +  1) * 4];
    c0 = WMMA_F32(a, b, c0);
    a = *(const v2f*)(px + k +  4); b.x = pb[(k +  4) * 4]; b.y = pb[(k +  5) * 4];
    c1 = WMMA_F32(a, b, c1);
    a = *(const v2f*)(px + k +  8); b.x = pb[(k +  8) * 4]; b.y = pb[(k +  9) * 4];
    c2 = WMMA_F32(a, b, c2);
    a = *(const v2f*)(px + k + 12); b.x = pb[(k + 12) * 4]; b.y = pb[(k + 13) * 4];
    c3 = WMMA_F32(a, b, c3);
  }

  // ---- phase 2: j in [256,1280) from hx ----
  const float* __restrict__ pb2 = pb + DIMX * 4;
  #pragma unroll 2
  for (int k = 0; k < DIMH; k += 16) {
    v2f a, b;
    a = *(const v2f*)(ph + k +  0); b.x = pb2[(k +  0) * 4]; b.y = pb2[(k +  1) * 4];
    c0 = WMMA_F32(a, b, c0);
    a = *(const v2f*)(ph + k +  4); b.x = pb2[(k +  4) * 4]; b.y = pb2[(k +  5) * 4];
    c1 = WMMA_F32(a, b, c1);
    a = *(const v2f*)(ph + k +  8); b.x = pb2[(k +  8) * 4]; b.y = pb2[(k +  9) * 4];
    c2 = WMMA_F32(a, b, c2);
    a = *(const v2f*)(ph + k + 12); b.x = pb2[(k + 12) * 4]; b.y = pb2[(k + 13) * 4];
    c3 = WMMA_F32(a, b, c3);
  }

  v8f c = (c0 + c1) + (c2 + c3);

  // D layout: VGPR v -> row (v + 8*half) of tile, N (=gate) = l15
  if (l15 < 4) {
    const int r0 = blockIdx.x * 16 + half * 8;
    #pragma unroll
    for (int v = 0; v < 8; ++v)
      a_ws[(size_t)(r0 + v) * 4 + l15] = c[v];
  }
}

// ---------------------------------------------------------------------------
// K2: per-row gate scalars.  <Z0> = cos(theta[0]) * cos(a + b[0])  (exact
// reduction of the 8-qubit circuit: RX/RY layers on |0..0> give a product
// state, and Z0 commutes with the CNOT chain -> only qubit 0 matters).
// Store {f, i*g, o} per row.
// ---------------------------------------------------------------------------
__global__ __launch_bounds__(256) void qlstm_gates(
    const float* __restrict__ a_ws,
    const float* __restrict__ bf, const float* __restrict__ bi,
    const float* __restrict__ bg, const float* __restrict__ bo,
    const float* __restrict__ thf, const float* __restrict__ thi,
    const float* __restrict__ thg, const float* __restrict__ tho,
    float4* __restrict__ g_ws) {
  const int b = blockIdx.x * blockDim.x + threadIdx.x;
  if (b >= BATCH) return;
  const float4 a = ((const float4*)a_ws)[b];
  const float evf = cosf(thf[0]) * cosf(a.x + bf[0]);
  const float evi = cosf(thi[0]) * cosf(a.y + bi[0]);
  const float evg = cosf(thg[0]) * cosf(a.z + bg[0]);
  const float evo = cosf(tho[0]) * cosf(a.w + bo[0]);
  const float fv = 1.0f / (1.0f + expf(-evf));
  const float iv = 1.0f / (1.0f + expf(-evi));
  const float gv = tanhf(evg);
  const float ov = 1.0f / (1.0f + expf(-evo));
  g_ws[b] = make_float4(fv, iv * gv, ov, 0.0f);
}

// ---------------------------------------------------------------------------
// K3: elementwise update, float4-vectorized (bandwidth bound: 64 MB in,
// 128 MB out -> dominates runtime at ~8.2 us @ 23.3 TB/s).
//   new_c = f*cx + i*g ;  new_h = o * tanh(new_c)
// out = [ new_h (B*H) | new_c (B*H) ]
// ---------------------------------------------------------------------------
__global__ __launch_bounds__(256) void qlstm_update(
    const float* __restrict__ cx, const float4* __restrict__ g_ws,
    float* __restrict__ out) {
  const int t = blockIdx.x * blockDim.x + threadIdx.x;   // t < B*H/4
  const int b = t >> 8;                                  // (t*4) / H, H=1024
  const float4 g  = g_ws[b];
  const float4 c4 = ((const float4*)cx)[t];
  float4 cn, hn;
  cn.x = fmaf(g.x, c4.x, g.y); hn.x = g.z * tanhf(cn.x);
  cn.y = fmaf(g.x, c4.y, g.y); hn.y = g.z * tanhf(cn.y);
  cn.z = fmaf(g.x, c4.z, g.y); hn.z = g.z * tanhf(cn.z);
  cn.w = fmaf(g.x, c4.w, g.y); hn.w = g.z * tanhf(cn.w);
  ((float4*)out)[t] = hn;
  ((float4*)out)[(size_t)(BATCH * DIMH / 4) + t] = cn;
}

extern "C" void kernel_launch(void* const* d_in, const int* in_sizes, int n_in,
                              void* d_out, int out_size, void* d_ws, size_t ws_size,
                              hipStream_t stream) {
  (void)in_sizes; (void)n_in; (void)out_size; (void)ws_size;
  const float* x   = (const float*)d_in[0];
  const float* hx  = (const float*)d_in[1];
  const float* cx  = (const float*)d_in[2];
  const float* Wf  = (const float*)d_in[3];
  const float* bf  = (const float*)d_in[4];
  const float* Wi  = (const float*)d_in[5];
  const float* bi  = (const float*)d_in[6];
  const float* Wg  = (const float*)d_in[7];
  const float* bg  = (const float*)d_in[8];
  const float* Wo  = (const float*)d_in[9];
  const float* bo  = (const float*)d_in[10];
  const float* thf = (const float*)d_in[11];
  const float* thi = (const float*)d_in[12];
  const float* thg = (const float*)d_in[13];
  const float* tho = (const float*)d_in[14];
  float* out = (float*)d_out;

  float*  a_ws = (float*)d_ws;                                               // B*4 f32
  float4* g_ws = (float4*)((char*)d_ws + (size_t)BATCH * 4 * sizeof(float)); // B f4

  qlstm_gemv_wmma<<<BATCH / 16, 32, 0, stream>>>(x, hx, Wf, Wi, Wg, Wo, a_ws);
  qlstm_gates<<<BATCH / 256, 256, 0, stream>>>(a_ws, bf, bi, bg, bo,
                                               thf, thi, thg, tho, g_ws);
  qlstm_update<<<(BATCH * DIMH / 4) / 256, 256, 0, stream>>>(cx, g_ws, out);
}